// BSA_25039659336180
// MI455X (gfx1250) — compile-verified
//
#include <hip/hip_runtime.h>
#include <hip/hip_bf16.h>
#include <math.h>

// ---------------------------------------------------------------------------
// BSA pipeline for MI455X (gfx1250, wave32).
// Memory-bound: ~0.6 GB total traffic @ 23.3 TB/s => ~26us floor.
// Covariance GEMM (per-batch 256x256, K=400, fp32) uses V_WMMA_F32_16X16X4_F32.
// ---------------------------------------------------------------------------

typedef __attribute__((ext_vector_type(2))) float v2f;
typedef __attribute__((ext_vector_type(8))) float v8f;

#define B_  16
#define C_  256
#define H_  128
#define W_  128
#define PLANE (H_ * W_)
#define POOL 20
#define NPOOL (POOL * POOL)   // 400
#define MID 16
#define OC_ 32
#define OH_ 64
#define OW_ 64

__device__ __forceinline__ float sigmoidf(float v) {
    return 1.0f / (1.0f + __expf(-v));
}

// ---------------------------------------------------------------------------
// 1) Adaptive avg-pool 128x128 -> 20x20 per (b,c) plane, plus per-(b,c) mean
//    over the 400 pooled values (needed for covariance centering).
//    One block per (b,c). Separable: pool W into LDS, then pool H.
// ---------------------------------------------------------------------------
__global__ void k_pool(const float* __restrict__ x,
                       float* __restrict__ xp,   // (B,C,400)
                       float* __restrict__ mu)   // (B,C)
{
    const int bc = blockIdx.x;
    const int tid = threadIdx.x;
    const float* xin = x + (size_t)bc * PLANE;

    __shared__ float rowp[H_ * POOL];   // 128 x 20
    __shared__ float outp[NPOOL];
    __shared__ float red[256];

    // Phase 1: pool along W. torch bins: s=floor(j*n/out), e=ceil((j+1)*n/out)
    for (int task = tid; task < H_ * POOL; task += 256) {
        int r = task / POOL, j = task % POOL;
        int s = (j * H_) / POOL;
        int e = ((j + 1) * H_ + POOL - 1) / POOL;
        const float* row = xin + (size_t)r * W_;
        float acc = 0.0f;
        for (int w = s; w < e; ++w) acc += row[w];
        rowp[task] = acc / (float)(e - s);
    }
    __syncthreads();

    // Phase 2: pool along H.
    for (int task = tid; task < NPOOL; task += 256) {
        int i = task / POOL, j = task % POOL;
        int s = (i * H_) / POOL;
        int e = ((i + 1) * H_ + POOL - 1) / POOL;
        float acc = 0.0f;
        for (int r = s; r < e; ++r) acc += rowp[r * POOL + j];
        float v = acc / (float)(e - s);
        outp[task] = v;
        xp[(size_t)bc * NPOOL + task] = v;
    }
    __syncthreads();

    // Mean over the 400 pooled values.
    float p = 0.0f;
    for (int t = tid; t < NPOOL; t += 256) p += outp[t];
    red[tid] = p;
    __syncthreads();
    for (int st = 128; st > 0; st >>= 1) {
        if (tid < st) red[tid] += red[tid + st];
        __syncthreads();
    }
    if (tid == 0) mu[bc] = red[0] / (float)NPOOL;
}

// ---------------------------------------------------------------------------
// 2) Per-batch covariance cov[b] = vc^T vc / 399 via V_WMMA_F32_16X16X4_F32.
//    vc[n,c] = xp[b,c,n] - mu[b,c].  One wave32 per 16x16 output tile.
//    xp is 6.5 MB => resident in the 192 MB L2, so the strided operand
//    gathers (channel stride 1600B) are L2 hits.
//    A-frag (16x4 f32): lanes 0-15 K={0,1}, lanes 16-31 K={2,3} (ISA 7.12.2).
//    D-frag: VGPR r -> M = r + 8*(lane>=16), N = lane%16.
// ---------------------------------------------------------------------------
__global__ void k_cov_wmma(const float* __restrict__ xp,
                           const float* __restrict__ mu,
                           float* __restrict__ cov)   // (B,C,C)
{
    const int b    = blockIdx.x;
    const int wave = threadIdx.x >> 5;          // 8 waves per block
    const int lane = threadIdx.x & 31;
    const int tile = blockIdx.y * 8 + wave;     // 0..255
    const int tm = tile >> 4, tn = tile & 15;
    const int half = lane >> 4, lr = lane & 15;

    const float* xpb = xp + (size_t)b * C_ * NPOOL;
    const float* mub = mu + b * C_;

    const int rowA = tm * 16 + lr;              // A channel (M)
    const int rowB = tn * 16 + lr;              // B channel (N)
    const float* pa = xpb + (size_t)rowA * NPOOL;
    const float* pb = xpb + (size_t)rowB * NPOOL;
    const float muA = mub[rowA];
    const float muB = mub[rowB];

    v8f c = {};
    for (int k0 = 0; k0 < NPOOL; k0 += 4) {
        const int k = k0 + half * 2;
        v2f a, bb;
        a.x  = pa[k]     - muA;
        a.y  = pa[k + 1] - muA;
        bb.x = pb[k]     - muB;
        bb.y = pb[k + 1] - muB;
        // D = A(16x4) * B(4x16) + C
        c = __builtin_amdgcn_wmma_f32_16x16x4_f32(
                /*neg_a=*/false, a, /*neg_b=*/false, bb,
                /*c_mod=*/(short)0, c, /*reuse_a=*/false, /*reuse_b=*/false);
    }

    const float scale = 1.0f / 399.0f;          // / (n-1)
    float* cb = cov + (size_t)b * C_ * C_;
#pragma unroll
    for (int r = 0; r < 8; ++r) {
        const int m = tm * 16 + r + 8 * half;
        const int n = tn * 16 + lr;
        cb[(size_t)m * C_ + n] = c[r] * scale;
    }
}

// ---------------------------------------------------------------------------
// 3) feat[b,c] = mean over d of cov[b,c,d]
// ---------------------------------------------------------------------------
__global__ void k_feat(const float* __restrict__ cov, float* __restrict__ feat)
{
    const int b = blockIdx.x, cch = blockIdx.y;
    const float* row = cov + ((size_t)b * C_ + cch) * C_;
    __shared__ float red[64];
    float s = 0.0f;
    for (int d = threadIdx.x; d < C_; d += 64) s += row[d];
    red[threadIdx.x] = s;
    __syncthreads();
    for (int st = 32; st > 0; st >>= 1) {
        if (threadIdx.x < st) red[threadIdx.x] += red[threadIdx.x + st];
        __syncthreads();
    }
    if (threadIdx.x == 0) feat[b * C_ + cch] = red[0] / (float)C_;
}

// ---------------------------------------------------------------------------
// 4) attn = sigmoid(feat @ W^T + bias)
// ---------------------------------------------------------------------------
__global__ void k_attn(const float* __restrict__ feat,
                       const float* __restrict__ W,
                       const float* __restrict__ bias,
                       float* __restrict__ attn)
{
    const int b = blockIdx.x;
    const int c = threadIdx.x;
    __shared__ float f[C_];
    f[c] = feat[b * C_ + c];
    __syncthreads();
    const float* wr = W + (size_t)c * C_;
    float s = bias[c];
    for (int d = 0; d < C_; ++d) s += f[d] * wr[d];
    attn[b * C_ + c] = sigmoidf(s);
}

// ---------------------------------------------------------------------------
// 5) score = attn.mean(b); stable top-16 (descending, ties -> lower index),
//    selected indices emitted in ascending channel order (== jnp.sort of
//    argsort[:16]).  Single block, fully deterministic.
// ---------------------------------------------------------------------------
__global__ void k_topk(const float* __restrict__ attn,
                       int* __restrict__ max_id,   // (16)
                       float* __restrict__ gain)   // (16) = 1 + score
{
    __shared__ float sc[C_];
    __shared__ int rk[C_];
    const int c = threadIdx.x;
    float s = 0.0f;
    for (int b = 0; b < B_; ++b) s += attn[b * C_ + c];
    s *= (1.0f / (float)B_);
    sc[c] = s;
    __syncthreads();
    int r = 0;
    for (int d = 0; d < C_; ++d) {
        float o = sc[d];
        if (o > s || (o == s && d < c)) ++r;
    }
    rk[c] = r;
    __syncthreads();
    if (r < MID) {
        int pos = 0;
        for (int d = 0; d < c; ++d) if (rk[d] < MID) ++pos;
        max_id[pos] = c;
        gain[pos]   = 1.0f + s;
    }
}

// ---------------------------------------------------------------------------
// 6) Fused big pass over x: build xc = [x1 | x2]  (b,32,h,w) and the
//    channel mean/max planes for LSA.  One thread per pixel, streaming all
//    256 channel planes coalesced; prefetch next plane (global_prefetch_b8).
// ---------------------------------------------------------------------------
__global__ void k_xc(const float* __restrict__ x,
                     const int* __restrict__ max_id,
                     const float* __restrict__ gain,
                     float* __restrict__ xc,      // (B,32,H,W)
                     float* __restrict__ smean,   // (B,H,W)
                     float* __restrict__ smax)    // (B,H,W)
{
    __shared__ int   mid[MID];
    __shared__ float gn[MID];
    if (threadIdx.x < MID) {
        mid[threadIdx.x] = max_id[threadIdx.x];
        gn[threadIdx.x]  = gain[threadIdx.x];
    }
    __syncthreads();

    const size_t pix = (size_t)blockIdx.x * blockDim.x + threadIdx.x; // b*16384+hw
    const int b  = (int)(pix >> 14);
    const int hw = (int)(pix & (PLANE - 1));
    const float* xb  = x  + (size_t)b * C_  * PLANE + hw;
    float*       xcb = xc + (size_t)b * OC_ * PLANE + hw;

    float mean = 0.0f;
    float mx   = -INFINITY;
    int p = 0;
    for (int i = 0; i < MID; ++i) {          // x2 group i
        float gs = 0.0f;
        for (int j = 0; j < 16; ++j) {
            const int cch = i * 16 + j;
            const float v = xb[(size_t)cch * PLANE];
            if (cch + 1 < C_)
                __builtin_prefetch(xb + (size_t)(cch + 1) * PLANE, 0, 1);
            gs += v;
            if (p < MID && mid[p] == cch) {  // selected channel -> x1[p]
                const float xv = v * gn[p];
                xcb[(size_t)p * PLANE] = xv;
                mean += xv;
                mx = fmaxf(mx, xv);
                ++p;
            }
        }
        const float g = gs * (1.0f / 16.0f);
        xcb[(size_t)(MID + i) * PLANE] = g;
        mean += g;
        mx = fmaxf(mx, g);
    }
    smean[pix] = mean * (1.0f / (float)OC_);
    smax[pix]  = mx;
}

// ---------------------------------------------------------------------------
// 7) LSA: 7x7 conv on [mean;max] (pad 3, no bias), sigmoid.
// ---------------------------------------------------------------------------
__global__ void k_lsa(const float* __restrict__ smean,
                      const float* __restrict__ smax,
                      const float* __restrict__ lw,   // (1,2,7,7)
                      float* __restrict__ sig)        // (B,H,W)
{
    __shared__ float w[98];
    if (threadIdx.x < 98) w[threadIdx.x] = lw[threadIdx.x];
    __syncthreads();

    const size_t pix = (size_t)blockIdx.x * blockDim.x + threadIdx.x;
    const int b  = (int)(pix >> 14);
    const int hw = (int)(pix & (PLANE - 1));
    const int yy = hw >> 7, xx = hw & (W_ - 1);
    const float* pm = smean + (size_t)b * PLANE;
    const float* px = smax  + (size_t)b * PLANE;

    float a = 0.0f;
    for (int kh = 0; kh < 7; ++kh) {
        const int iy = yy + kh - 3;
        if (iy < 0 || iy >= H_) continue;
        for (int kw = 0; kw < 7; ++kw) {
            const int ix = xx + kw - 3;
            if (ix < 0 || ix >= W_) continue;
            const int idx = iy * W_ + ix;
            a += pm[idx] * w[kh * 7 + kw] + px[idx] * w[49 + kh * 7 + kw];
        }
    }
    sig[pix] = sigmoidf(a);
}

// ---------------------------------------------------------------------------
// 8) CBA conv: 3x3, stride 2, pad 1, on xa = xc * sig.  LDS-tiled:
//    each block computes an 8x8 output tile for all 32 oc; the 17x17x32
//    input tile (37 KB) is staged in LDS with the sigmoid gate applied.
// ---------------------------------------------------------------------------
#define TIN 17
__global__ void k_cba(const float* __restrict__ xc,
                      const float* __restrict__ sig,
                      const float* __restrict__ cw,   // (32,32,3,3)
                      const float* __restrict__ cb,   // (32)
                      float* __restrict__ y)          // (B,32,64,64)
{
    __shared__ float t[OC_ * TIN * TIN];   // 9248 floats = 37 KB
    const int b  = blockIdx.x;
    const int oh0 = blockIdx.y * 8;
    const int ow0 = blockIdx.z * 8;
    const int ih0 = oh0 * 2 - 1;
    const int iw0 = ow0 * 2 - 1;
    const float* xcb = xc  + (size_t)b * OC_ * PLANE;
    const float* sb  = sig + (size_t)b * PLANE;

    for (int idx = threadIdx.x; idx < OC_ * TIN * TIN; idx += 256) {
        const int ic  = idx / (TIN * TIN);
        const int rem = idx % (TIN * TIN);
        const int r   = rem / TIN, cc = rem % TIN;
        const int ih  = ih0 + r, iw = iw0 + cc;
        float v = 0.0f;
        if (ih >= 0 && ih < H_ && iw >= 0 && iw < W_) {
            const int pidx = ih * W_ + iw;
            v = xcb[(size_t)ic * PLANE + pidx] * sb[pidx];
        }
        t[idx] = v;
    }
    __syncthreads();

    const int pixid = threadIdx.x & 63;     // 64 output pixels
    const int ocg   = threadIdx.x >> 6;     // 4 groups x 8 oc
    const int po = pixid >> 3, pw = pixid & 7;

    float acc[8];
#pragma unroll
    for (int i = 0; i < 8; ++i) acc[i] = cb[ocg * 8 + i];

    for (int ic = 0; ic < OC_; ++ic) {
        const float* tb = &t[ic * TIN * TIN + (po * 2) * TIN + pw * 2];
#pragma unroll
        for (int kh = 0; kh < 3; ++kh) {
#pragma unroll
            for (int kw = 0; kw < 3; ++kw) {
                const float v = tb[kh * TIN + kw];
#pragma unroll
                for (int i = 0; i < 8; ++i) {
                    const int oc = ocg * 8 + i;
                    acc[i] += v * cw[((oc * OC_ + ic) * 9) + kh * 3 + kw];
                }
            }
        }
    }

    const int oh = oh0 + po, ow = ow0 + pw;
#pragma unroll
    for (int i = 0; i < 8; ++i) {
        const int oc = ocg * 8 + i;
        y[(((size_t)b * OC_ + oc) * OH_ + oh) * OW_ + ow] = acc[i];
    }
}

// ---------------------------------------------------------------------------
// 9) BN batch stats per output channel (deterministic tree reduction).
// ---------------------------------------------------------------------------
__global__ void k_bnstat(const float* __restrict__ y,
                         float* __restrict__ bnmu, float* __restrict__ bnrs)
{
    const int oc = blockIdx.x;
    __shared__ float rs[256], rq[256];
    float s = 0.0f, q = 0.0f;
    for (int b = 0; b < B_; ++b) {
        const float* p = y + ((size_t)b * OC_ + oc) * (OH_ * OW_);
        for (int i = threadIdx.x; i < OH_ * OW_; i += 256) {
            const float v = p[i];
            s += v;
            q += v * v;
        }
    }
    rs[threadIdx.x] = s;
    rq[threadIdx.x] = q;
    __syncthreads();
    for (int st = 128; st > 0; st >>= 1) {
        if (threadIdx.x < st) {
            rs[threadIdx.x] += rs[threadIdx.x + st];
            rq[threadIdx.x] += rq[threadIdx.x + st];
        }
        __syncthreads();
    }
    if (threadIdx.x == 0) {
        const float inv_n = 1.0f / (float)(B_ * OH_ * OW_);
        const float m   = rs[0] * inv_n;
        const float var = rq[0] * inv_n - m * m;
        bnmu[oc] = m;
        bnrs[oc] = rsqrtf(var + 1e-3f);
    }
}

// ---------------------------------------------------------------------------
// 10) BN apply + SiLU -> d_out
// ---------------------------------------------------------------------------
__global__ void k_bnapply(const float* __restrict__ y,
                          const float* __restrict__ bnmu,
                          const float* __restrict__ bnrs,
                          const float* __restrict__ gamma,
                          const float* __restrict__ beta,
                          float* __restrict__ out)
{
    const size_t i = (size_t)blockIdx.x * blockDim.x + threadIdx.x;
    const int oc = (int)((i >> 12) & (OC_ - 1));   // / (64*64) % 32
    const float yn = (y[i] - bnmu[oc]) * bnrs[oc] * gamma[oc] + beta[oc];
    out[i] = yn * sigmoidf(yn);
}

// ---------------------------------------------------------------------------
extern "C" void kernel_launch(void* const* d_in, const int* in_sizes, int n_in,
                              void* d_out, int out_size, void* d_ws, size_t ws_size,
                              hipStream_t stream)
{
    const float* x        = (const float*)d_in[0];
    const float* linear_w = (const float*)d_in[1];
    const float* linear_b = (const float*)d_in[2];
    const float* lsa_w    = (const float*)d_in[3];
    const float* conv_w   = (const float*)d_in[4];
    const float* conv_b   = (const float*)d_in[5];
    const float* bn_gamma = (const float*)d_in[6];
    const float* bn_beta  = (const float*)d_in[7];

    // Workspace layout (~56 MB total), 256B-aligned slices.
    char* ws = (char*)d_ws;
    size_t off = 0;
    auto take = [&](size_t bytes) -> void* {
        void* p = ws + off;
        off += (bytes + 255) & ~(size_t)255;
        return p;
    };
    float* xp    = (float*)take((size_t)B_ * C_ * NPOOL * 4);   // 6.55 MB
    float* mu    = (float*)take((size_t)B_ * C_ * 4);
    float* cov   = (float*)take((size_t)B_ * C_ * C_ * 4);      // 4 MB
    float* feat  = (float*)take((size_t)B_ * C_ * 4);
    float* attn  = (float*)take((size_t)B_ * C_ * 4);
    int*   maxid = (int*)  take(MID * 4);
    float* gainv = (float*)take(MID * 4);
    float* xc    = (float*)take((size_t)B_ * OC_ * PLANE * 4);  // 33.5 MB
    float* smean = (float*)take((size_t)B_ * PLANE * 4);        // 1 MB
    float* smax  = (float*)take((size_t)B_ * PLANE * 4);        // 1 MB
    float* sig   = (float*)take((size_t)B_ * PLANE * 4);        // 1 MB
    float* ybuf  = (float*)take((size_t)B_ * OC_ * OH_ * OW_ * 4); // 8.4 MB
    float* bnmu  = (float*)take(OC_ * 4);
    float* bnrs  = (float*)take(OC_ * 4);
    (void)ws_size; (void)in_sizes; (void)n_in; (void)out_size;

    // 1. pool + per-(b,c) pooled mean
    k_pool<<<dim3(B_ * C_), 256, 0, stream>>>(x, xp, mu);
    // 2. covariance GEMM via V_WMMA_F32_16X16X4_F32 (8 waves/block, 1 tile/wave)
    k_cov_wmma<<<dim3(B_, 32), 256, 0, stream>>>(xp, mu, cov);
    // 3. feat = cov row mean
    k_feat<<<dim3(B_, C_), 64, 0, stream>>>(cov, feat);
    // 4. attn = sigmoid(feat @ W^T + b)
    k_attn<<<dim3(B_), 256, 0, stream>>>(feat, linear_w, linear_b, attn);
    // 5. deterministic top-16 channel selection
    k_topk<<<dim3(1), 256, 0, stream>>>(attn, maxid, gainv);
    // 6. fused xc build + channel mean/max (second full pass over x)
    k_xc<<<dim3((B_ * PLANE) / 256), 256, 0, stream>>>(x, maxid, gainv, xc, smean, smax);
    // 7. 7x7 spatial-attention conv + sigmoid
    k_lsa<<<dim3((B_ * PLANE) / 256), 256, 0, stream>>>(smean, smax, lsa_w, sig);
    // 8. gated 3x3 stride-2 conv (LDS-tiled)
    k_cba<<<dim3(B_, OH_ / 8, OW_ / 8), 256, 0, stream>>>(xc, sig, conv_w, conv_b, ybuf);
    // 9. BN batch stats
    k_bnstat<<<dim3(OC_), 256, 0, stream>>>(ybuf, bnmu, bnrs);
    // 10. BN apply + SiLU -> out
    k_bnapply<<<dim3((B_ * OC_ * OH_ * OW_) / 256), 256, 0, stream>>>(
        ybuf, bnmu, bnrs, bn_gamma, bn_beta, (float*)d_out);
}